// Net_9560597201379
// MI455X (gfx1250) — compile-verified
//
#include <hip/hip_runtime.h>

// ---------------------------------------------------------------------------
// SNN encoder/decoder for MI455X (gfx1250, wave32).
// All GEMMs via V_WMMA_F32_16X16X4_F32 (exact fp32 accumulate).
// One wave == one 32x32 macro-tile (2x2 WMMA tiles): each loaded A/B fragment
// feeds two WMMAs, halving L2 traffic vs one-tile-per-wave. EXEC is all-1s
// for every WMMA (guards are wave-uniform; M%32==0, N%32==0, K%4==0).
// ---------------------------------------------------------------------------

typedef __attribute__((ext_vector_type(2))) float v2f;
typedef __attribute__((ext_vector_type(8))) float v8f;

#define BETA 0.9f

// WMMA f32 16x16x4 VGPR layout (ISA 7.12.2, 32-bit operands):
//   A(m,k): lane = (m&15) + 16*(k>>1), vgpr = k&1  -> per-lane float2 load
//   B(k,n): lane = (n&15) + 16*(k>>1), vgpr = k&1  -> per-lane float2 load
//   C(m,n): vgpr = m%8,  lane = (n&15) + 16*(m/8)
__device__ __forceinline__ v8f wmma4(v2f a, v2f b, v8f c) {
  return __builtin_amdgcn_wmma_f32_16x16x4_f32(
      /*neg_a=*/false, a, /*neg_b=*/false, b,
      /*c_mod=*/(short)0, c, /*reuse_a=*/false, /*reuse_b=*/false);
}

// Accumulate a 32x32 macro-tile of A[M,K] @ W[N,K]^T into acc[2][2].
// m = this lane's row inside M-tile 0, n = this lane's col inside N-tile 0,
// khalf = 2*(lane>>4).
__device__ __forceinline__ void wmma_tile_2x2(const float* __restrict__ A,
                                              const float* __restrict__ W,
                                              int K, int m, int n, int khalf,
                                              v8f acc[2][2]) {
  const float* a0 = A + (size_t)m * K + khalf;        // M-tile 0
  const float* a1 = a0 + (size_t)16 * K;              // M-tile 1
  const float* w0 = W + (size_t)n * K + khalf;        // N-tile 0
  const float* w1 = w0 + (size_t)16 * K;              // N-tile 1
#pragma unroll 2
  for (int k = 0; k < K; k += 4) {
    v2f fa0 = *(const v2f*)(a0 + k);
    v2f fa1 = *(const v2f*)(a1 + k);
    v2f fb0 = *(const v2f*)(w0 + k);
    v2f fb1 = *(const v2f*)(w1 + k);
    acc[0][0] = wmma4(fa0, fb0, acc[0][0]);
    acc[0][1] = wmma4(fa0, fb1, acc[0][1]);
    acc[1][0] = wmma4(fa1, fb0, acc[1][0]);
    acc[1][1] = wmma4(fa1, fb1, acc[1][1]);
  }
}

// C[M,N] = A[M,K] @ W[N,K]^T + bias[N]
__global__ __launch_bounds__(256) void gemm_bias(
    const float* __restrict__ A, const float* __restrict__ W,
    const float* __restrict__ bias, float* __restrict__ C,
    int M, int N, int K) {
  int wave = (int)((blockIdx.x * 256u + threadIdx.x) >> 5);
  int lane = (int)(threadIdx.x & 31u);
  int ntiles = N >> 5;                    // 32-wide macro-tiles
  int mt = wave / ntiles;
  int nt = wave - mt * ntiles;
  if (mt >= (M >> 5)) return;             // wave-uniform guard

  int lm = lane & 15;
  int lh = lane >> 4;
  int m = mt * 32 + lm;
  int n = nt * 32 + lm;

  v8f acc[2][2] = {{{}, {}}, {{}, {}}};
  wmma_tile_2x2(A, W, K, m, n, lh * 2, acc);

  float bn[2] = {bias[n], bias[n + 16]};
  int r0 = mt * 32 + lh * 8;
#pragma unroll
  for (int mi = 0; mi < 2; ++mi)
#pragma unroll
    for (int ni = 0; ni < 2; ++ni)
#pragma unroll
      for (int r = 0; r < 8; ++r)
        C[(size_t)(r0 + mi * 16 + r) * N + n + ni * 16] = acc[mi][ni][r] + bn[ni];
}

// cur = A @ W^T + b, then snntorch Leaky (reset='subtract') on mstate:
//   reset = (m_old > th) ? th : 0
//   m_new = BETA*m_old + cur - reset
//   s     = (m_new > th) ? 1 : 0
// Optionally streams s / m_new to extra destinations (decoder output slices).
__global__ __launch_bounds__(256) void gemm_bias_lif(
    const float* __restrict__ A, const float* __restrict__ W,
    const float* __restrict__ bias, float* __restrict__ mstate,
    float* __restrict__ sout, float* __restrict__ out_m,
    float* __restrict__ out_s, int M, int N, int K, float thresh) {
  int wave = (int)((blockIdx.x * 256u + threadIdx.x) >> 5);
  int lane = (int)(threadIdx.x & 31u);
  int ntiles = N >> 5;
  int mt = wave / ntiles;
  int nt = wave - mt * ntiles;
  if (mt >= (M >> 5)) return;

  int lm = lane & 15;
  int lh = lane >> 4;
  int m = mt * 32 + lm;
  int n = nt * 32 + lm;

  v8f acc[2][2] = {{{}, {}}, {{}, {}}};
  wmma_tile_2x2(A, W, K, m, n, lh * 2, acc);

  float bn[2] = {bias[n], bias[n + 16]};
  int r0 = mt * 32 + lh * 8;
#pragma unroll
  for (int mi = 0; mi < 2; ++mi)
#pragma unroll
    for (int ni = 0; ni < 2; ++ni)
#pragma unroll
      for (int r = 0; r < 8; ++r) {
        size_t idx = (size_t)(r0 + mi * 16 + r) * N + n + ni * 16;
        float cur = acc[mi][ni][r] + bn[ni];
        float mo = mstate[idx];
        float mn = BETA * mo + cur - ((mo > thresh) ? thresh : 0.0f);
        float s = (mn > thresh) ? 1.0f : 0.0f;
        mstate[idx] = mn;
        if (sout) sout[idx] = s;
        if (out_m) out_m[idx] = mn;
        if (out_s) out_s[idx] = s;
      }
}

// Standalone LIF (layers 1 and 3): update membrane, emit spikes.
__global__ __launch_bounds__(256) void lif_fire(
    float* __restrict__ mstate, const float* __restrict__ cur,
    float* __restrict__ sout, int n, float thresh) {
  int i = (int)(blockIdx.x * 256u + threadIdx.x);
  if (i >= n) return;
  float mo = mstate[i];
  float mn = BETA * mo + cur[i] - ((mo > thresh) ? thresh : 0.0f);
  mstate[i] = mn;
  sout[i] = (mn > thresh) ? 1.0f : 0.0f;
}

static inline void launch_gemm(const float* A, const float* W, const float* b,
                               float* C, int M, int N, int K,
                               hipStream_t stream) {
  int tiles = (M >> 5) * (N >> 5);
  int blocks = (tiles + 7) / 8;  // 8 waves (macro-tiles) per 256-thread block
  gemm_bias<<<blocks, 256, 0, stream>>>(A, W, b, C, M, N, K);
}

static inline void launch_gemm_lif(const float* A, const float* W,
                                   const float* b, float* mstate, float* sout,
                                   float* out_m, float* out_s, int M, int N,
                                   int K, float thresh, hipStream_t stream) {
  int tiles = (M >> 5) * (N >> 5);
  int blocks = (tiles + 7) / 8;
  gemm_bias_lif<<<blocks, 256, 0, stream>>>(A, W, b, mstate, sout, out_m,
                                            out_s, M, N, K, thresh);
}

extern "C" void kernel_launch(void* const* d_in, const int* in_sizes, int n_in,
                              void* d_out, int out_size, void* d_ws,
                              size_t ws_size, hipStream_t stream) {
  (void)in_sizes; (void)n_in; (void)out_size; (void)ws_size;

  const float* x  = (const float*)d_in[0];  // [8,128,512]
  const float* W1 = (const float*)d_in[1];  // [256,512]
  const float* b1 = (const float*)d_in[2];
  const float* W2 = (const float*)d_in[3];  // [128,256]
  const float* b2 = (const float*)d_in[4];
  const float* W3 = (const float*)d_in[5];  // [256,128]
  const float* b3 = (const float*)d_in[6];
  const float* W4 = (const float*)d_in[7];  // [512,256]
  const float* b4 = (const float*)d_in[8];

  float* out = (float*)d_out;
  const size_t HALF = 8ull * 8 * 8 * 128 * 512;  // 33554432 (mem | spk split)

  // Workspace partition (floats). Membrane states first -> single memset.
  const size_t SZ_M1 = 1024ull * 256;   // encoder L1 mem
  const size_t SZ_M2 = 1024ull * 128;   // encoder L2 mem
  const size_t SZ_M3 = 8192ull * 256;   // decoder L1 mem
  const size_t SZ_M4 = 8192ull * 512;   // decoder L2 mem
  const size_t SZ_CUR1 = 1024ull * 256;
  const size_t SZ_S1 = 1024ull * 256;
  const size_t SZ_SPK = 8ull * 1024 * 128;  // spk_rec [8,8,128,128]
  const size_t SZ_CUR3 = 8192ull * 256;

  float* ws = (float*)d_ws;
  float* m1 = ws;
  float* m2 = m1 + SZ_M1;
  float* m3 = m2 + SZ_M2;
  float* m4 = m3 + SZ_M3;
  float* cur1 = m4 + SZ_M4;
  float* s1 = cur1 + SZ_CUR1;
  float* spk = s1 + SZ_S1;
  float* cur3 = spk + SZ_SPK;
  float* s3 = cur3 + SZ_CUR3;

  // Zero all membrane state (graph-capture-safe async memset).
  hipMemsetAsync(d_ws, 0, (SZ_M1 + SZ_M2 + SZ_M3 + SZ_M4) * sizeof(float),
                 stream);

  // ---- Encoder ----
  // cur1 = x @ W1^T + b1 : [1024,512]x[512,256] -> [1024,256]
  launch_gemm(x, W1, b1, cur1, 1024, 256, 512, stream);

  for (int t = 0; t < 8; ++t) {
    // m1/s1 update (thresh=1)
    lif_fire<<<(int)(SZ_M1 / 256), 256, 0, stream>>>(m1, cur1, s1,
                                                     (int)SZ_M1, 1.0f);
    // cur2 = s1 @ W2^T + b2, fused m2-LIF, spikes -> spk_rec[t]
    launch_gemm_lif(s1, W2, b2, m2, spk + (size_t)t * 1024 * 128,
                    nullptr, nullptr, 1024, 128, 256, 1.0f, stream);
  }

  // ---- Decoder ----
  // cur3 = spk_rec @ W3^T + b3 : [8192,128]x[128,256] -> [8192,256]
  launch_gemm(spk, W3, b3, cur3, 8192, 256, 128, stream);

  for (int t = 0; t < 8; ++t) {
    // m3/s3 update (thresh=1)
    lif_fire<<<(int)(SZ_M3 / 256), 256, 0, stream>>>(m3, cur3, s3,
                                                     (int)SZ_M3, 1.0f);
    // cur4 = s3 @ W4^T + b4, fused m4-LIF (thresh=20000),
    // stream m4 -> mem_rec_1[t], s4 -> spk_rec_1[t]
    launch_gemm_lif(s3, W4, b4, m4, nullptr,
                    out + (size_t)t * 8192 * 512,
                    out + HALF + (size_t)t * 8192 * 512,
                    8192, 512, 256, 20000.0f, stream);
  }
}